// Group_19593640804465
// MI455X (gfx1250) — compile-verified
//
#include <hip/hip_runtime.h>

// Problem constants from the reference.
#define B_  32
#define N_  8192
#define D_  14
#define G_  512
#define K_  32

typedef __attribute__((ext_vector_type(2))) float v2f;
typedef __attribute__((ext_vector_type(8))) float v8f;

// ---------------------------------------------------------------------------
// Kernel 1: Furthest Point Sampling. One workgroup (1024 threads = 32 waves)
// per batch. Points packed as float4 (x,y,z,0) in LDS so the critical loop
// issues 8x ds_load_b128 instead of 24x ds_load_b32 per thread per step.
// Running min-distance kept in registers (8 floats/thread). Each of the 511
// steps does a two-level wave32 argmax (lexicographic: max value, min index
// on ties, matching jnp.argmax). Finally writes center[b,g,0:14].
// ---------------------------------------------------------------------------
__global__ __launch_bounds__(1024) void fps_kernel(const float* __restrict__ xyz,
                                                   float* __restrict__ center) {
  extern __shared__ char smem[];
  float4* pts   = (float4*)smem;            // N_ (128 KB)
  float* redV   = (float*)(pts + N_);       // 32
  int*   redI   = (int*)(redV + 32);        // 32
  int*   idxbuf = redI + 32;                // G_
  int*   bI     = idxbuf + G_;              // 1

  const int tid  = threadIdx.x;
  const int b    = blockIdx.x;
  const int wid  = tid >> 5;
  const int lane = tid & 31;
  const size_t xb = (size_t)b * N_ * D_;

  for (int i = tid; i < N_; i += 1024) {
    const float* p = xyz + xb + (size_t)i * D_;
    float2 xy = *(const float2*)p;          // row stride 56B -> 8B aligned
    pts[i] = make_float4(xy.x, xy.y, p[2], 0.0f);
  }
  if (tid == 0) idxbuf[0] = 0;              // reference starts at index 0
  __syncthreads();

  float dst[8];
#pragma unroll
  for (int u = 0; u < 8; ++u) dst[u] = 1.0e10f;   // matches reference init

  float4 lp = pts[0];

  for (int s = 1; s < G_; ++s) {
    float bestv = -1.0f; int besti = N_;
#pragma unroll
    for (int u = 0; u < 8; ++u) {
      int i = tid + u * 1024;
      float4 p = pts[i];                    // ds_load_b128, conflict-free
      float dx = p.x - lp.x, dy = p.y - lp.y, dz = p.z - lp.z;
      float d  = dx * dx + dy * dy + dz * dz;
      float nd = fminf(dst[u], d);
      dst[u] = nd;
      if (nd > bestv) { bestv = nd; besti = i; }   // strict > keeps lowest i
    }
    // wave32 butterfly argmax (value desc, index asc on ties)
#pragma unroll
    for (int off = 16; off > 0; off >>= 1) {
      float ov = __shfl_xor(bestv, off, 32);
      int   oi = __shfl_xor(besti, off, 32);
      if (ov > bestv || (ov == bestv && oi < besti)) { bestv = ov; besti = oi; }
    }
    if (lane == 0) { redV[wid] = bestv; redI[wid] = besti; }
    __syncthreads();
    if (wid == 0) {
      float v = redV[lane]; int i2 = redI[lane];   // exactly 32 wave winners
#pragma unroll
      for (int off = 16; off > 0; off >>= 1) {
        float ov = __shfl_xor(v, off, 32);
        int   oi = __shfl_xor(i2, off, 32);
        if (ov > v || (ov == v && oi < i2)) { v = ov; i2 = oi; }
      }
      if (lane == 0) { idxbuf[s] = i2; *bI = i2; }
    }
    __syncthreads();
    lp = pts[*bI];                          // LDS broadcast read
  }
  __syncthreads();

  // center[b,g,:] = xyz[b, idx[g], :]  (all 14 dims, 7x float2)
  for (int g = tid; g < G_; g += 1024) {
    int n = idxbuf[g];
    const float* src = xyz + xb + (size_t)n * D_;
    float* dstp = center + ((size_t)b * G_ + g) * D_;
#pragma unroll
    for (int d = 0; d < D_; d += 2)
      *(float2*)(dstp + d) = *(const float2*)(src + d);
  }
}

// ---------------------------------------------------------------------------
// Kernel 2: fused score (WMMA f32 16x16x4) + top-32 selection + gather.
// One workgroup (256 threads = 8 waves) per (batch, 16-group tile).
// Score s[m,n] = |x_n|^2 - 2*c_m.x_n  (same ranking as d2; row-constant |c|^2
// dropped), computed as a single V_WMMA_F32_16X16X4_F32 per 16x16 tile with
//   A = [-2cx, -2cy, -2cz, 1]  (16x4),  B = [x, y, z, |x|^2]^T (4x16).
// A-layout: lanes 0-15 hold K0/K1, lanes 16-31 hold K2/K3 (ISA 7.12.2).
// Point loads are software-pipelined one step ahead (carried across quarter
// boundaries) so global_load_b96 latency hides under WMMA + ds stores and the
// selection phase. Scores staged in LDS in 4 quarters of 16x2048; each wave
// owns 2 rows and extracts the 32 smallest in ascending (value,index) order
// via ds_load_b128 sweeps + wave32 lexicographic reduction + INF marking.
// 4 sorted runs per row are merged, then neighbors gathered and re-centered.
// ---------------------------------------------------------------------------
__global__ __launch_bounds__(256) void group_kernel(const float* __restrict__ xyz,
                                                    const float* __restrict__ center,
                                                    float* __restrict__ nb) {
  extern __shared__ char smem[];
  float* scores = (float*)smem;             // 16*2048 floats (128 KB)
  float* candV  = scores + 16 * 2048;       // 16*128
  int*   candC  = (int*)(candV + 16 * 128); // 16*128
  int*   topC   = candC + 16 * 128;         // 16*32

  const int tid   = threadIdx.x;
  const int lane  = tid & 31;
  const int wave  = tid >> 5;
  const int gbase = blockIdx.x * 16;
  const int b     = blockIdx.y;
  const size_t xb = (size_t)b * N_ * D_;
  const float INFv = __builtin_inff();
  const int colw  = wave * 16 + (lane & 15);   // this lane's column slot

  // Build WMMA A operand (constant for this workgroup).
  const int m_a = lane & 15;
  const size_t cb = ((size_t)b * G_ + gbase + m_a) * D_;
  float cx = center[cb + 0], cy = center[cb + 1], cz = center[cb + 2];
  v2f A;
  A[0] = (lane < 16) ? (-2.0f * cx) : (-2.0f * cz);   // K0 | K2
  A[1] = (lane < 16) ? (-2.0f * cy) : 1.0f;           // K1 | K3

  // Software-pipelined point loader: chunk t in [0,64), 128 columns each.
  auto load_pt = [&](int t, float2& xy, float& z) {
    int p = t * 128 + colw;
    const float* pp = xyz + xb + (size_t)p * D_;
    __builtin_prefetch(pp + 256 * D_, 0, 1);          // speculative stream-ahead
    xy = *(const float2*)pp;
    z  = pp[2];
  };

  float2 cur_xy; float cur_z;
  load_pt(0, cur_xy, cur_z);

  for (int q = 0; q < 4; ++q) {
    // ---- scoring: columns q*2048 .. q*2048+2047, 8 waves x 16 cols/step ----
    for (int stp = 0; stp < 16; ++stp) {
      int t = q * 16 + stp;
      float2 nxt_xy; float nxt_z;
      load_pt((t + 1 < 64) ? t + 1 : 0, nxt_xy, nxt_z);   // issue next load early

      float w2 = cur_xy.x * cur_xy.x + cur_xy.y * cur_xy.y + cur_z * cur_z;
      v2f Bm;
      Bm[0] = (lane < 16) ? cur_xy.x : cur_z;         // K0 row | K2 row
      Bm[1] = (lane < 16) ? cur_xy.y : w2;            // K1 row | K3 row
      v8f acc = {};
      acc = __builtin_amdgcn_wmma_f32_16x16x4_f32(
          /*neg_a=*/false, A, /*neg_b=*/false, Bm,
          /*c_mod=*/(short)0, acc, /*reuse_a=*/false, /*reuse_b=*/false);
      int colq = stp * 128 + colw;
      int mb = (lane >> 4) * 8;                       // rows 0-7 | 8-15
#pragma unroll
      for (int r = 0; r < 8; ++r)
        scores[(mb + r) * 2048 + colq] = acc[r];

      cur_xy = nxt_xy; cur_z = nxt_z;
    }
    __syncthreads();

    // ---- selection: each wave owns rows wave*2 + {0,1} ----
    for (int rw = 0; rw < 2; ++rw) {
      int m = wave * 2 + rw;
      const float4* rowp = (const float4*)(scores + m * 2048);
      for (int k = 0; k < K_; ++k) {
        float bv = INFv; int bc = 0x7fffffff;
#pragma unroll
        for (int j = 0; j < 16; ++j) {
          float4 v = rowp[j * 32 + lane];             // ds_load_b128, no conflicts
          int c = (j * 32 + lane) * 4;
          if (v.x < bv) { bv = v.x; bc = c + 0; }     // ascending c: strict <
          if (v.y < bv) { bv = v.y; bc = c + 1; }     //   keeps lowest index
          if (v.z < bv) { bv = v.z; bc = c + 2; }
          if (v.w < bv) { bv = v.w; bc = c + 3; }
        }
#pragma unroll
        for (int off = 16; off > 0; off >>= 1) {      // wave32 lex-min butterfly
          float ov = __shfl_xor(bv, off, 32);
          int   oc = __shfl_xor(bc, off, 32);
          if (ov < bv || (ov == bv && oc < bc)) { bv = ov; bc = oc; }
        }
        if (lane == 0) {
          candV[m * 128 + q * 32 + k] = bv;
          candC[m * 128 + q * 32 + k] = q * 2048 + bc;
          scores[m * 2048 + bc] = INFv;               // mark extracted
        }
      }
    }
    __syncthreads();
  }

  // ---- 4-way merge of sorted runs per row (one thread per row) ----
  if (tid < 16) {
    int m = tid;
    int h0 = 0, h1 = 0, h2 = 0, h3 = 0;
    for (int k = 0; k < K_; ++k) {
      float bestv = INFv; int bestc = 0x7fffffff; int bq = 0;
      if (h0 < 32) { float v = candV[m*128 +   0 + h0]; int c = candC[m*128 +   0 + h0];
        if (v < bestv || (v == bestv && c < bestc)) { bestv = v; bestc = c; bq = 0; } }
      if (h1 < 32) { float v = candV[m*128 +  32 + h1]; int c = candC[m*128 +  32 + h1];
        if (v < bestv || (v == bestv && c < bestc)) { bestv = v; bestc = c; bq = 1; } }
      if (h2 < 32) { float v = candV[m*128 +  64 + h2]; int c = candC[m*128 +  64 + h2];
        if (v < bestv || (v == bestv && c < bestc)) { bestv = v; bestc = c; bq = 2; } }
      if (h3 < 32) { float v = candV[m*128 +  96 + h3]; int c = candC[m*128 +  96 + h3];
        if (v < bestv || (v == bestv && c < bestc)) { bestv = v; bestc = c; bq = 3; } }
      topC[m * K_ + k] = bestc;
      if (bq == 0) h0++; else if (bq == 1) h1++; else if (bq == 2) h2++; else h3++;
    }
  }
  __syncthreads();

  // ---- gather neighbors, subtract center from dims 0..2 ----
  for (int pk = tid; pk < 16 * K_; pk += 256) {
    int m = pk >> 5, k = pk & 31;
    int n = topC[m * K_ + k];
    const float* src = xyz + xb + (size_t)n * D_;
    size_t crow = ((size_t)b * G_ + gbase + m) * D_;
    float* dstp = nb + (((size_t)b * G_ + gbase + m) * K_ + k) * D_;
    float c0 = center[crow + 0], c1 = center[crow + 1], c2 = center[crow + 2];
    float2 t;
    t = *(const float2*)(src + 0); t.x -= c0; t.y -= c1; *(float2*)(dstp + 0) = t;
    t = *(const float2*)(src + 2); t.x -= c2;            *(float2*)(dstp + 2) = t;
#pragma unroll
    for (int d = 4; d < D_; d += 2)
      *(float2*)(dstp + d) = *(const float2*)(src + d);
  }
}

// ---------------------------------------------------------------------------
extern "C" void kernel_launch(void* const* d_in, const int* in_sizes, int n_in,
                              void* d_out, int out_size, void* d_ws, size_t ws_size,
                              hipStream_t stream) {
  (void)in_sizes; (void)n_in; (void)out_size; (void)d_ws; (void)ws_size;
  const float* xyz = (const float*)d_in[0];
  float* out    = (float*)d_out;
  float* nb     = out;                                   // (B,G,K,D) flat
  float* center = out + (size_t)B_ * G_ * K_ * D_;       // (B,G,D) flat

  size_t lds1 = (size_t)N_ * 16 + 32 * 4 + 32 * 4 + G_ * 4 + 16;            // ~130.3 KB
  size_t lds2 = (size_t)16 * 2048 * 4 + 16 * 128 * 4 * 2 + 16 * K_ * 4;     // ~146 KB

  (void)hipFuncSetAttribute((const void*)fps_kernel,
                            hipFuncAttributeMaxDynamicSharedMemorySize, (int)lds1);
  (void)hipFuncSetAttribute((const void*)group_kernel,
                            hipFuncAttributeMaxDynamicSharedMemorySize, (int)lds2);

  fps_kernel<<<B_, 1024, lds1, stream>>>(xyz, center);
  group_kernel<<<dim3(G_ / 16, B_), 256, lds2, stream>>>(xyz, center, nb);
}